// CTCComputationLayer_70738111365586
// MI455X (gfx1250) — compile-verified
//
#include <hip/hip_runtime.h>
#include <cstdint>

#define CTC_EPS 1e-7f
#define CTC_VEPS (1024.0f * 1e-7f)   // V * EPS added to the row sum

typedef __attribute__((ext_vector_type(2))) float v2f;
typedef __attribute__((ext_vector_type(8))) float v8f;

// ---------------------------------------------------------------------------
// Kernel 1: per 16 rows (b,t): denom = sum_v(p)+V*eps via WMMA row-sum of an
// async-staged LDS tile; gather blank + 64 label columns; emit ratios
// (p_id+eps)/denom  ->  ws[(b*T+t)*rstride + slot], slot0=blank, slot1+j=label j.
// ---------------------------------------------------------------------------
__global__ __launch_bounds__(32) void ctc_rowprep(
    const float* __restrict__ probs,   // [B*T, 1024]
    const int*   __restrict__ labels,  // [B, 64]
    float*       __restrict__ ratios,  // [B*T, rstride]
    int rstride)
{
  __shared__ float smem[16 * 516 + 16];      // 16 rows x 516 pitch + 16 row sums
  const int lane = threadIdx.x;
  const long long bt0 = (long long)blockIdx.x * 16;
  const int b = (int)(bt0 >> 9);             // 512 t's per b; 16 | 512 so uniform

  const int* lab = labels + b * 64;
  const int id0 = (lane == 0) ? 1023 : lab[lane - 1];  // slot  lane   (blank at slot 0)
  const int id1 = lab[lane + 31];                      // slot  lane+32
  const int id2 = lab[63];                             // slot  64 (lane 0 only)

  const uint32_t lds_base = (uint32_t)(uintptr_t)(&smem[0]);

  v8f acc0 = {0.f, 0.f, 0.f, 0.f, 0.f, 0.f, 0.f, 0.f};
  v8f acc1 = acc0;
  const v2f ones = {1.0f, 1.0f};

  float g0[16] = {}, g1[16] = {}, g2[16] = {};

  const int row  = lane & 15;
  const int koff = (lane >> 4) << 1;         // A-fragment K offset per ISA layout

  for (int h = 0; h < 2; ++h) {              // two half-rows of 512 floats
    // ---- async DMA: 16 rows x 2KB into LDS (pitch 516 floats) ----
    const char* gsrc = (const char*)probs +
        ((size_t)bt0 * 1024 + (size_t)h * 512) * 4 + (size_t)lane * 16;
    #pragma unroll
    for (int r = 0; r < 16; ++r) {
      uint64_t ga = (uint64_t)(uintptr_t)(gsrc + (size_t)r * 4096);
      uint32_t la = lds_base + (uint32_t)(r * (516 * 4) + lane * 16);
      asm volatile(
        "global_load_async_to_lds_b128 %0, %1, off\n\t"
        "global_load_async_to_lds_b128 %0, %1, off offset:512\n\t"
        "global_load_async_to_lds_b128 %0, %1, off offset:1024\n\t"
        "global_load_async_to_lds_b128 %0, %1, off offset:1536"
        :: "v"(la), "v"(ga) : "memory");
    }
    asm volatile("s_wait_asynccnt 0" ::: "memory");
    __syncthreads();

    // ---- WMMA row sums: D = A(16x4) * ones(4x16) accumulated over K ----
    const float* tp = &smem[row * 516 + koff];
    #pragma unroll 4
    for (int k = 0; k < 512; k += 8) {
      v2f a0 = *(const v2f*)(tp + k);
      v2f a1 = *(const v2f*)(tp + k + 4);
      acc0 = __builtin_amdgcn_wmma_f32_16x16x4_f32(false, a0, false, ones,
                                                   (short)0, acc0, false, false);
      acc1 = __builtin_amdgcn_wmma_f32_16x16x4_f32(false, a1, false, ones,
                                                   (short)0, acc1, false, false);
    }

    // ---- gather this lane's label columns while tile is resident ----
    if ((id0 >> 9) == h) {
      #pragma unroll
      for (int m = 0; m < 16; ++m) g0[m] = smem[m * 516 + (id0 & 511)];
    }
    if ((id1 >> 9) == h) {
      #pragma unroll
      for (int m = 0; m < 16; ++m) g1[m] = smem[m * 516 + (id1 & 511)];
    }
    if (lane == 0 && (id2 >> 9) == h) {
      #pragma unroll
      for (int m = 0; m < 16; ++m) g2[m] = smem[m * 516 + (id2 & 511)];
    }
    __syncthreads();   // before next half overwrites the tile
  }

  // ---- publish row sums (C/D layout: lanes 0-15 hold M=0..7, 16-31 M=8..15)
  float* sums = &smem[16 * 516];
  if (lane == 0) {
    #pragma unroll
    for (int r = 0; r < 8; ++r) sums[r] = acc0[r] + acc1[r];
  }
  if (lane == 16) {
    #pragma unroll
    for (int r = 0; r < 8; ++r) sums[8 + r] = acc0[r] + acc1[r];
  }
  __syncthreads();

  // ---- emit ratios ----
  #pragma unroll
  for (int m = 0; m < 16; ++m) {
    const float inv = 1.0f / (sums[m] + CTC_VEPS);
    float* o = ratios + (size_t)(bt0 + m) * rstride;
    o[lane]      = (g0[m] + CTC_EPS) * inv;
    o[lane + 32] = (g1[m] + CTC_EPS) * inv;
    if (lane == 0) o[64] = (g2[m] + CTC_EPS) * inv;
  }
}

// ---------------------------------------------------------------------------
// Kernel 2: scaled CTC forward recursion, one wave per batch.
// State s = 32*i + lane, i = 0..4 (i==4 valid only for lane 0: s = 128).
// alpha kept in LDS; per-step renorm; nll = -(sum log c_t + log(aT[S-2]+aT[S-1]))
// ---------------------------------------------------------------------------
__global__ __launch_bounds__(32) void ctc_alpha(
    const float* __restrict__ ratios,
    const int*   __restrict__ labels,
    float*       __restrict__ out,
    int rstride)
{
  __shared__ float aL[132];
  const int b = blockIdx.x;
  const int lane = threadIdx.x;
  const bool odd = (lane & 1) != 0;
  const int* lab = labels + b * 64;

  int slot0 = 0;
  bool allow[4] = {false, false, false, false};
  if (odd) {
    const int j0 = (lane - 1) >> 1;
    slot0 = 1 + j0;
    #pragma unroll
    for (int i = 0; i < 4; ++i) {
      const int j = j0 + 16 * i;                // label index of state s = 32i+lane
      allow[i] = (j > 0) && (lab[j] != lab[j - 1]);
    }
  }

  const float* rbase = ratios + (size_t)b * 512 * rstride;

  // ---- t = 0: alpha0[0] = p(blank), alpha0[1] = p(label0) ----
  float n0 = 0.0f;
  if (lane == 0) n0 = rbase[0];
  if (lane == 1) n0 = rbase[1];
  float c = n0;
  #pragma unroll
  for (int off = 16; off > 0; off >>= 1) c += __shfl_xor(c, off, 32);
  float logsum = __logf(c);
  {
    const float inv = 1.0f / c;
    #pragma unroll
    for (int i = 0; i < 4; ++i) aL[32 * i + lane] = (i == 0) ? n0 * inv : 0.0f;
    if (lane == 0) aL[128] = 0.0f;
  }
  __syncthreads();

  for (int t = 1; t < 512; ++t) {
    const float* rt = rbase + (size_t)t * rstride;
    const float rb = rt[0];                    // blank emission ratio
    float rl[4];
    if (odd) {
      #pragma unroll
      for (int i = 0; i < 4; ++i) rl[i] = rt[slot0 + 16 * i];
    } else {
      rl[0] = rl[1] = rl[2] = rl[3] = rb;
    }

    float n[5];
    #pragma unroll
    for (int i = 0; i < 4; ++i) {
      const int s = 32 * i + lane;
      const float a1 = aL[s];
      const float a2 = (s > 0) ? aL[s - 1] : 0.0f;
      const float a3 = (odd && allow[i]) ? aL[s - 2] : 0.0f;
      n[i] = (a1 + a2 + a3) * rl[i];
    }
    n[4] = (lane == 0) ? (aL[128] + aL[127]) * rb : 0.0f;   // s = 128 (blank)

    float cs = n[0] + n[1] + n[2] + n[3] + n[4];
    #pragma unroll
    for (int off = 16; off > 0; off >>= 1) cs += __shfl_xor(cs, off, 32);
    logsum += __logf(cs);
    const float cinv = 1.0f / cs;

    __syncthreads();
    #pragma unroll
    for (int i = 0; i < 4; ++i) aL[32 * i + lane] = n[i] * cinv;
    if (lane == 0) aL[128] = n[4] * cinv;
    __syncthreads();
  }

  if (lane == 0) out[b] = -(logsum + __logf(aL[127] + aL[128]));
}

// ---------------------------------------------------------------------------
extern "C" void kernel_launch(void* const* d_in, const int* in_sizes, int n_in,
                              void* d_out, int out_size, void* d_ws, size_t ws_size,
                              hipStream_t stream) {
  const int*   labels = (const int*)d_in[0];     // [256, 64] int32
  const float* probs  = (const float*)d_in[1];   // [256, 512, 1024] f32
  float* out    = (float*)d_out;                 // [256, 1] f32
  float* ratios = (float*)d_ws;

  const size_t BT = 256 * 512;
  int rstride = 68;                               // padded (272B rows)
  if (ws_size < BT * 68 * sizeof(float)) rstride = 66;
  if (ws_size < BT * 66 * sizeof(float)) rstride = 65;

  ctc_rowprep<<<(int)(BT / 16), 32, 0, stream>>>(probs, labels, ratios, rstride);
  ctc_alpha<<<256, 32, 0, stream>>>(ratios, labels, out, rstride);
}